// ResidualVQ_3169685864518
// MI455X (gfx1250) — compile-verified
//
#include <hip/hip_runtime.h>

typedef float v2f __attribute__((ext_vector_type(2)));
typedef float v8f __attribute__((ext_vector_type(8)));

#define NUM_Q 8
#define KCB   1024
#define DDIM  256
#define NVEC  65536
#define ROWS_PER_WAVE 16
#define WAVES_PER_BLOCK 8

// LDS tile: 32 codewords x 256 floats, row stride padded to 1040B
// (16B-aligned for async b128 writes; rows 4 banks apart -> conflict-free
//  16-row ds_load_b64 B-fragment reads)
#define ROW_BYTES   1040u
#define ROW_FLOATS  260
#define TILE_FLOATS (32 * ROW_FLOATS)            // 8320 floats = 33280 B
#define NBUF        3
#define LDS_BYTES   (NBUF * 32 * ROW_BYTES)      // 99840 B
#define NTILES      (NUM_Q * KCB / 32)           // 256 tiles of 32 codewords

// ---------------------------------------------------------------------------
// Kernel 1: wnorm[q][k] = sum_d W[q][k][d]^2
// ---------------------------------------------------------------------------
__global__ void vq_wnorm_kernel(const float* __restrict__ cb,
                                float* __restrict__ wnorm) {
    int e = blockIdx.x * blockDim.x + threadIdx.x;   // 0 .. 8191
    if (e >= NUM_Q * KCB) return;
    const float4* p = (const float4*)(cb + (size_t)e * DDIM);
    float s = 0.f;
#pragma unroll 8
    for (int i = 0; i < DDIM / 4; ++i) {
        float4 v = p[i];
        s += v.x * v.x + v.y * v.y + v.z * v.z + v.w * v.w;
    }
    wnorm[e] = s;
}

// ---------------------------------------------------------------------------
// Kernel 2: zero the scalar loss slot (d_out is poisoned; loss uses atomicAdd)
// ---------------------------------------------------------------------------
__global__ void vq_zero_loss(float* __restrict__ out) {
    if (threadIdx.x == 0 && blockIdx.x == 0)
        out[(size_t)NVEC * DDIM + (size_t)NVEC * NUM_Q] = 0.f;
}

// ---------------------------------------------------------------------------
// Async staging of one 32-codeword tile (32 KB) into an LDS buffer.
// Tile t = bytes [t*32768, t*32768+32768) of the codebook (contiguous
// across quantizers). Each thread issues 8 GLOBAL_LOAD_ASYNC_TO_LDS_B128
// (GVS mode: saddr = codebook base, vaddr = 32-bit offset, vdst = LDS addr).
// Per wave: 8 async ops -> ASYNCcnt bookkeeping below.
// ---------------------------------------------------------------------------
__device__ __forceinline__ void issue_tile(const float* cbBase,
                                           unsigned ldsBase, int t, int tid) {
    const unsigned ldsBuf = ldsBase + (unsigned)(t % NBUF) * (32u * ROW_BYTES);
    const unsigned glBase = (unsigned)t * 32768u;
#pragma unroll
    for (int c = 0; c < 8; ++c) {
        unsigned chunk    = (unsigned)(tid + c * 256);       // 0..2047 (16B units)
        unsigned b        = chunk * 16u;                     // byte in packed tile
        unsigned lds_addr = ldsBuf + b + (chunk >> 6) * 16u; // +16B pad per row
        unsigned gl_off   = glBase + b;
        asm volatile("global_load_async_to_lds_b128 %0, %1, %2"
                     :: "v"(lds_addr), "v"(gl_off), "s"(cbBase)
                     : "memory");
    }
}

// ---------------------------------------------------------------------------
// Kernel 3: main residual-VQ kernel.
//   One wave owns 16 rows; residual lives in VGPRs in the exact f32
//   A-fragment layout of V_WMMA_F32_16X16X4_F32. Codeword tiles are staged
//   block-cooperatively into LDS via the async engine (triple-buffered),
//   and B-fragments are read with ds_load_b64.
// ---------------------------------------------------------------------------
__global__ void __launch_bounds__(WAVES_PER_BLOCK * 32, 1)
vq_main_kernel(const float* __restrict__ z,
               const float* __restrict__ cb,
               const float* __restrict__ wnorm,
               float* __restrict__ out) {
    extern __shared__ float sh[];                      // NBUF tile buffers

    const int tid    = threadIdx.x;
    const int lane   = tid & 31;
    const int wave   = tid >> 5;
    const int sub    = lane & 15;       // residual row owned by this lane
    const int half   = lane >> 4;       // which K-pair of the fragment
    const int colOff = 2 * half;        // column offset within each 4-group
    const int n0     = (blockIdx.x * WAVES_PER_BLOCK + wave) * ROWS_PER_WAVE;
    const unsigned ldsBase = (unsigned)(size_t)&sh[0]; // low 32 bits = LDS addr

    // ---- residual := z (A-fragment layout, register resident) ----
    const float* zrow = z + (size_t)(n0 + sub) * DDIM + colOff;
    v2f res[64];
#pragma unroll
    for (int j = 0; j < 64; ++j)
        res[j] = *(const v2f*)(zrow + 4 * j);

    float lossAcc = 0.f;
    float* outIdx  = out + (size_t)NVEC * DDIM;
    float* outLoss = out + (size_t)NVEC * DDIM + (size_t)NVEC * NUM_Q;

    // ---- prime the async pipeline with tiles 0 and 1 ----
    issue_tile(cb, ldsBase, 0, tid);
    issue_tile(cb, ldsBase, 1, tid);

    for (int q = 0; q < NUM_Q; ++q) {
        const float* Wq  = cb + (size_t)q * KCB * DDIM;
        const float* wnq = wnorm + q * KCB;

        float bestD[8];
        int   bestK[8];
#pragma unroll
        for (int i = 0; i < 8; ++i) { bestD[i] = 3.4e38f; bestK[i] = 0; }

        for (int ct = 0; ct < KCB / 32; ++ct) {
            const int t = q * (KCB / 32) + ct;

            // Tile t done when <=8 of my async ops remain (tile t+1 in flight);
            // last tile has nothing behind it -> drain fully.
            if (t == NTILES - 1) {
                asm volatile("s_wait_asynccnt 0x0" ::: "memory");
            } else {
                asm volatile("s_wait_asynccnt 0x8" ::: "memory");
            }
            __syncthreads();   // publish all waves' chunks of tile t;
                               // also: tile t-1 readers are done -> safe to
                               // overwrite buffer (t+2)%3 below.
            if (t + 2 < NTILES) issue_tile(cb, ldsBase, t + 2, tid);

            const float* bufp = sh + (size_t)(t % NBUF) * TILE_FLOATS;
            const float* b0p  = bufp + sub * ROW_FLOATS + colOff;        // rows 0..15
            const float* b1p  = b0p + 16 * ROW_FLOATS;                    // rows 16..31

            v8f acc0 = {};
            v8f acc1 = {};
#pragma unroll
            for (int j = 0; j < 64; ++j) {          // K(D) step of 4
                v2f a  = res[j];
                v2f b0 = *(const v2f*)(b0p + 4 * j);
                v2f b1 = *(const v2f*)(b1p + 4 * j);
                acc0 = __builtin_amdgcn_wmma_f32_16x16x4_f32(
                    false, a, false, b0, (short)0, acc0, false, false);
                acc1 = __builtin_amdgcn_wmma_f32_16x16x4_f32(
                    false, a, false, b1, (short)0, acc1, false, false);
            }

            // distance (up to row-constant ||r||^2): ||W_k||^2 - 2 r.W_k
            const int k0 = ct * 32 + sub;
            const int k1 = k0 + 16;
            const float wn0 = wnq[k0];
            const float wn1 = wnq[k1];
#pragma unroll
            for (int i = 0; i < 8; ++i) {
                float d0 = wn0 - 2.0f * acc0[i];
                float d1 = wn1 - 2.0f * acc1[i];
                if (d0 < bestD[i]) { bestD[i] = d0; bestK[i] = k0; }
                if (d1 < bestD[i]) { bestD[i] = d1; bestK[i] = k1; }
            }
        }

        // ---- argmin across the 16 lanes of each half (N dimension) ----
#pragma unroll
        for (int i = 0; i < 8; ++i) {
#pragma unroll
            for (int m = 1; m < 16; m <<= 1) {
                float od = __shfl_xor(bestD[i], m, 32);
                int   ok = __shfl_xor(bestK[i], m, 32);
                if (od < bestD[i] || (od == bestD[i] && ok < bestK[i])) {
                    bestD[i] = od; bestK[i] = ok;
                }
            }
        }

        // vk = bestK[lane&7] (unrolled select; every lane supplies its slot)
        int vk = bestK[0];
#pragma unroll
        for (int i = 1; i < 8; ++i) vk = ((lane & 7) == i) ? bestK[i] : vk;

        // store indices: lane with sub<8 in half h owns row sub + 8h
        if (sub < 8) {
            int row = sub + 8 * half;
            outIdx[(size_t)(n0 + row) * NUM_Q + q] = (float)vk;
        }

        // broadcast winning codeword for this lane's residual row (= sub)
        int src = (sub < 8) ? sub : (sub + 8);
        int myk = __shfl(vk, src, 32);

        // ---- residual -= W[myk]; accumulate new-residual^2 for the loss ----
        const float* wk = Wq + (size_t)myk * DDIM + colOff;
        float sq = 0.f;
#pragma unroll
        for (int j = 0; j < 64; ++j) {
            v2f w = *(const v2f*)(wk + 4 * j);
            res[j] -= w;
            sq += res[j][0] * res[j][0] + res[j][1] * res[j][1];
        }
        lossAcc += sq;
    }

    // ---- z_reconstructed = z - final_residual ----
    float* orow = out + (size_t)(n0 + sub) * DDIM + colOff;
#pragma unroll
    for (int j = 0; j < 64; ++j) {
        v2f zv = *(const v2f*)(zrow + 4 * j);
        v2f r  = zv - res[j];
        *(v2f*)(orow + 4 * j) = r;
    }

    // ---- total_loss = 1.25 * sum(res_q^2) / (N*D), wave-reduced + atomicAdd --
    lossAcc *= (1.25f / ((float)NVEC * (float)DDIM));
#pragma unroll
    for (int off = 16; off > 0; off >>= 1)
        lossAcc += __shfl_down(lossAcc, off, 32);
    if (lane == 0) atomicAdd(outLoss, lossAcc);
}

// ---------------------------------------------------------------------------
extern "C" void kernel_launch(void* const* d_in, const int* in_sizes, int n_in,
                              void* d_out, int out_size, void* d_ws, size_t ws_size,
                              hipStream_t stream) {
    const float* z  = (const float*)d_in[0];                 // [N, D]
    const float* cb = (const float*)d_in[1];                 // [Q, K, D]
    float* out   = (float*)d_out;                            // recon | idx | loss
    float* wnorm = (float*)d_ws;                             // 8192 floats

    (void)in_sizes; (void)n_in; (void)out_size; (void)ws_size;

    vq_wnorm_kernel<<<(NUM_Q * KCB + 255) / 256, 256, 0, stream>>>(cb, wnorm);
    vq_zero_loss<<<1, 64, 0, stream>>>(out);

    const int blocks = NVEC / (WAVES_PER_BLOCK * ROWS_PER_WAVE);   // 512
    vq_main_kernel<<<blocks, WAVES_PER_BLOCK * 32, LDS_BYTES, stream>>>(
        z, cb, wnorm, out);
}